// FilterDetections_88862873354784
// MI455X (gfx1250) — compile-verified
//
#include <hip/hip_runtime.h>

// FilterDetections for MI455X (gfx1250, wave32).
// boxes: (8, 65536, 4) f32 ; classification: (8, 65536, 20) f32
// out: boxes (8,100,4) f32 | scores (8,100) f32 | labels (8,100) i32, flat-concat.

#define NB   8
#define NN   65536
#define NC   20
#define MD   100
#define TPB  1024
#define CAND (NC * MD)       // 2000 candidates per image

#define HAS_ASYNC_LDS                                                 \
  (defined(__AMDGCN__) &&                                             \
   __has_builtin(__builtin_amdgcn_global_load_async_to_lds_b32) &&    \
   __has_builtin(__builtin_amdgcn_s_wait_asynccnt))

__device__ __forceinline__ float neg_inf_f() { return __int_as_float((int)0xFF800000); }

__device__ __forceinline__ unsigned long long umax64(unsigned long long a,
                                                     unsigned long long b) {
  return a > b ? a : b;
}

// ---------------------------------------------------------------------------
// Kernel 1: one workgroup per (batch, class). Greedy NMS, 100 steps.
// All 65536 scores live in LDS (256 KB of the WGP's 320 KB) — element i is
// owned exclusively by thread (i % TPB), so score reads/writes need no
// barriers. Each step fuses "suppress vs previous winner" with "argmax of
// survivors" in a single pass; boxes stream from L2 (8.4 MB total, resident).
// Key encoding: (float_bits(score) << 32) | ~index  -> u64 max == (max score,
// lowest index) since valid scores are positive. Key 0 == nothing valid.
// ---------------------------------------------------------------------------
__global__ __launch_bounds__(TPB) void fd_nms_kernel(
    const float* __restrict__ boxes, const float* __restrict__ cls,
    float* __restrict__ cand_sc, int* __restrict__ cand_idx) {
  const int bc  = blockIdx.x;
  const int b   = bc / NC;
  const int c   = bc - b * NC;
  const int tid = threadIdx.x;

  const float*  clsp = cls + (size_t)b * NN * NC + c;
  const float4* bx   = (const float4*)(boxes + (size_t)b * NN * 4);

  __shared__ float              s[NN];     // 256 KB score array
  __shared__ unsigned long long red[TPB];  // 8 KB reduction tree

  // Stage scores global -> LDS on the CDNA5 async path; warm boxes into L2.
#if HAS_ASYNC_LDS
  for (int i = tid; i < NN; i += TPB) {
    __builtin_amdgcn_global_load_async_to_lds_b32(
        (__attribute__((address_space(1))) int*)(void*)(clsp + (size_t)i * NC),
        (__attribute__((address_space(3))) int*)(void*)(&s[i]), 0, 0);
    if ((i & 3) == 0) __builtin_prefetch((const void*)&bx[i], 0, 3);
  }
  __builtin_amdgcn_s_wait_asynccnt(0);
#else
  for (int i = tid; i < NN; i += TPB) {
    s[i] = clsp[(size_t)i * NC];
    if ((i & 3) == 0) __builtin_prefetch((const void*)&bx[i], 0, 3);
  }
#endif
  __syncthreads();

  float cbx0 = 0.f, cby0 = 0.f, cbx1 = 0.f, cby1 = 0.f, areaA = 0.f;
  bool haveBox = false;

  float* csc = cand_sc + (size_t)bc * MD;
  int*   cix = cand_idx + (size_t)bc * MD;

  for (int k = 0; k < MD; ++k) {
    unsigned long long key = 0ull;
    for (int i = tid; i < NN; i += TPB) {
      const float v = s[i];
      if (v > 0.01f) {                    // strict >, matches SCORE_THRESHOLD
        bool keep = true;
        if (haveBox) {
          const float4 bb = bx[i];
          // Exact IEEE f32, no FMA contraction (matches reference HLO math).
          const float ix1   = fmaxf(cbx0, bb.x);
          const float iy1   = fmaxf(cby0, bb.y);
          const float ix2   = fminf(cbx1, bb.z);
          const float iy2   = fminf(cby1, bb.w);
          const float iw    = fmaxf(__fsub_rn(ix2, ix1), 0.0f);
          const float ih    = fmaxf(__fsub_rn(iy2, iy1), 0.0f);
          const float inter = __fmul_rn(iw, ih);
          const float areaB = __fmul_rn(__fsub_rn(bb.z, bb.x), __fsub_rn(bb.w, bb.y));
          const float uni   = __fsub_rn(__fadd_rn(areaA, areaB), inter);
          const float iou   = (uni > 0.0f) ? __fdiv_rn(inter, uni) : 0.0f;
          if (iou > 0.5f) { s[i] = neg_inf_f(); keep = false; }
        }
        if (keep) {
          key = umax64(key, ((unsigned long long)__float_as_uint(v) << 32) |
                               (unsigned int)(~i));
        }
      }
    }

    // Block max-reduce of the u64 keys (LDS tree, wave32-safe).
    red[tid] = key;
    __syncthreads();
    for (int st = TPB >> 1; st > 0; st >>= 1) {
      if (tid < st) red[tid] = umax64(red[tid], red[tid + st]);
      __syncthreads();
    }
    const unsigned long long best = red[0];
    __syncthreads();  // protect red[] reuse next iteration

    if (best == 0ull) {  // nothing valid remains: reference emits -inf forever
      if (tid == 0) {
        for (int kk = k; kk < MD; ++kk) { csc[kk] = neg_inf_f(); cix[kk] = 0; }
      }
      break;  // uniform condition across block
    }

    const unsigned int sel = ~(unsigned int)(best & 0xffffffffull);
    if (tid == 0) {
      csc[k] = __uint_as_float((unsigned int)(best >> 32));
      cix[k] = (int)sel;
    }
    const float4 nb = bx[sel];  // uniform address -> broadcast load
    cbx0 = nb.x; cby0 = nb.y; cbx1 = nb.z; cby1 = nb.w;
    areaA = __fmul_rn(__fsub_rn(cbx1, cbx0), __fsub_rn(cby1, cby0));
    haveBox = true;  // winner itself gets suppressed next pass (IoU == 1)
  }
}

// ---------------------------------------------------------------------------
// Kernel 2: per image, global top-100 of 2000 (class, det) candidates.
// Candidate scores staged to LDS via the CDNA5 async-to-LDS path. Key low
// bits (CAND-1-t) => ties pick the lower flat index, matching lax.top_k.
// ---------------------------------------------------------------------------
__global__ __launch_bounds__(256) void fd_topk_kernel(
    const float* __restrict__ boxes, const float* __restrict__ cand_sc,
    const int* __restrict__ cand_idx, float* __restrict__ out_boxes,
    float* __restrict__ out_scores, int* __restrict__ out_labels) {
  const int b   = blockIdx.x;
  const int tid = threadIdx.x;

  __shared__ float              s_sc[CAND];
  __shared__ unsigned long long keys[CAND];
  __shared__ unsigned long long red[256];

  const float* scp = cand_sc + (size_t)b * CAND;

#if HAS_ASYNC_LDS
  for (int t = tid; t < CAND; t += 256) {
    __builtin_amdgcn_global_load_async_to_lds_b32(
        (__attribute__((address_space(1))) int*)(void*)(scp + t),
        (__attribute__((address_space(3))) int*)(void*)(&s_sc[t]), 0, 0);
  }
  __builtin_amdgcn_s_wait_asynccnt(0);
#else
  for (int t = tid; t < CAND; t += 256) s_sc[t] = scp[t];
#endif
  __syncthreads();

  for (int t = tid; t < CAND; t += 256) {
    const unsigned int bits = __float_as_uint(s_sc[t]);
    // Order-preserving f32 -> u32 transform (handles -inf markers).
    const unsigned int m = (bits & 0x80000000u) ? ~bits : (bits | 0x80000000u);
    keys[t] = ((unsigned long long)m << 32) | (unsigned int)(CAND - 1 - t);
  }
  __syncthreads();

  const float4* bx = (const float4*)(boxes + (size_t)b * NN * 4);

  for (int k = 0; k < MD; ++k) {
    unsigned long long key = 0ull;
    for (int t = tid; t < CAND; t += 256) key = umax64(key, keys[t]);
    red[tid] = key;
    __syncthreads();
    for (int st = 128; st > 0; st >>= 1) {
      if (tid < st) red[tid] = umax64(red[tid], red[tid + st]);
      __syncthreads();
    }
    if (tid == 0) {
      const unsigned long long best = red[0];
      const int   t     = CAND - 1 - (int)(best & 0xffffffffull);
      const float sc    = s_sc[t];
      const bool  valid = sc > neg_inf_f();
      const int   lab   = t / MD;
      const int   bi    = cand_idx[(size_t)b * CAND + t];
      float4 bb;
      if (valid) bb = bx[bi];
      else { bb.x = -1.0f; bb.y = -1.0f; bb.z = -1.0f; bb.w = -1.0f; }
      float* ob = out_boxes + ((size_t)b * MD + k) * 4;
      ob[0] = bb.x; ob[1] = bb.y; ob[2] = bb.z; ob[3] = bb.w;
      out_scores[(size_t)b * MD + k] = valid ? sc : -1.0f;
      out_labels[(size_t)b * MD + k] = valid ? lab : -1;
      keys[t] = 0ull;  // remove from further rounds
    }
    __syncthreads();
  }
}

extern "C" void kernel_launch(void* const* d_in, const int* in_sizes, int n_in,
                              void* d_out, int out_size, void* d_ws, size_t ws_size,
                              hipStream_t stream) {
  const float* boxes = (const float*)d_in[0];
  const float* cls   = (const float*)d_in[1];

  float* cand_sc  = (float*)d_ws;                                   // 64000 B
  int*   cand_idx = (int*)((char*)d_ws + (size_t)NB * CAND * sizeof(float));

  float* out        = (float*)d_out;
  float* out_boxes  = out;                        // 8*100*4 = 3200 floats
  float* out_scores = out + (size_t)NB * MD * 4;  // 800 floats
  int*   out_labels = (int*)(out + (size_t)NB * MD * 4 + (size_t)NB * MD);

  fd_nms_kernel<<<NB * NC, TPB, 0, stream>>>(boxes, cls, cand_sc, cand_idx);
  fd_topk_kernel<<<NB, 256, 0, stream>>>(boxes, cand_sc, cand_idx,
                                         out_boxes, out_scores, out_labels);
}